// Bridge_37220186587404
// MI455X (gfx1250) — compile-verified
//
#include <hip/hip_runtime.h>

typedef __attribute__((ext_vector_type(2))) float v2f;
typedef __attribute__((ext_vector_type(8))) float v8f;

#define EPSL     1e-08f
#define INV_TEMP 200.0f        /* 1 / (2*0.05^2) */
#define IDX_MASK 0x3FFFu       /* 14 bits: N = 16384 */
#define KEY_MASK 0xFFFFC000u

__device__ __forceinline__ unsigned umin32(unsigned a, unsigned b) { return a < b ? a : b; }
__device__ __forceinline__ unsigned umax32(unsigned a, unsigned b) { return a > b ? a : b; }

// ---- Pass 1: bake padded positions (x,y,z,0) and (u = 1/(w^2+eps), t = |p|^2*u) ----
__global__ void bake_kernel(const float* __restrict__ pos,
                            const float* __restrict__ wts,
                            float4* __restrict__ pos4,
                            float2* __restrict__ ut, int N) {
    int n = blockIdx.x * blockDim.x + threadIdx.x;
    if (n < N) {
        float x = pos[n * 3 + 0], y = pos[n * 3 + 1], z = pos[n * 3 + 2];
        float w = wts[n];
        float iw = 1.0f / (w * w + EPSL);
        float pn = x * x + y * y + z * z;
        float4 p; p.x = x; p.y = y; p.z = z; p.w = 0.0f;
        pos4[n] = p;
        float2 v; v.x = iw; v.y = pn * iw;
        ut[n] = v;
    }
}

// ---- Pass 2: WMMA distance tiles + per-lane packed top-8 + softmax feature gather ----
__global__ __launch_bounds__(256) void knn_interp_kernel(
        const float* __restrict__ coords,
        const float4* __restrict__ pos4,
        const float* __restrict__ feat,
        const float2* __restrict__ ut,
        float* __restrict__ out,
        int N) {
    __shared__ float lds_w[8][16][8];
    __shared__ int   lds_i[8][16][8];

    const int lane  = threadIdx.x & 31;
    const int wave  = threadIdx.x >> 5;
    const int qbase = blockIdx.x * 128 + wave * 16;
    const int qn    = lane & 15;       // query column / anchor row within tile
    const int h     = lane >> 4;       // K-half selector / anchor-half (m+8)
    const int q     = qbase + qn;

    // Query coordinate for this lane's column; forms B fragment (K x N = 4x16).
    const float cx = coords[q * 3 + 0];
    const float cy = coords[q * 3 + 1];
    const float cz = coords[q * 3 + 2];
    const float cnorm = cx * cx + cy * cy + cz * cz;

    v2f bfrag;                          // lanes 0-15: (K0,K1)=(x,y); lanes 16-31: (K2,K3)=(z,0)
    bfrag.x = h ? cz : cx;
    bfrag.y = h ? 0.0f : cy;

    v8f czero = {};

    // Sorted-ascending packed top-8 keys: (dist_bits & KEY_MASK) | anchor_idx
    unsigned s[8];
#pragma unroll
    for (int i = 0; i < 8; ++i) s[i] = 0xFFFFFFFFu;

    // Base pointer for this lane's A-fragment rows: padded float4 positions,
    // +8B for the upper half-wave so the b64 load grabs (z, 0) instead of (x, y).
    const char* apbase = (const char*)pos4 + h * 8;

    for (int base = 0; base < N; base += 16) {
        // A fragment (M x K = 16 anchors x 4): single uniform b64 load per lane.
        const int m = base + qn;
        const v2f afrag = *(const v2f*)(apbase + (size_t)m * 16);

        // (u,t) pairs for this lane's 8 candidate anchors: base+8h .. base+8h+7
        const float4* utp = (const float4*)(ut + base + 8 * h);
        const float4 g0 = utp[0], g1 = utp[1], g2 = utp[2], g3 = utp[3];

        // D[m][n] = pos[m] . coord[n]; lane holds 8 anchors (VGPR v -> m=base+8h+v)
        v8f d = __builtin_amdgcn_wmma_f32_16x16x4_f32(
            false, afrag, false, bfrag, (short)0, czero, false, false);

#pragma unroll
        for (int v = 0; v < 8; v += 2) {
            const float4 g = (v < 2) ? g0 : (v < 4) ? g1 : (v < 6) ? g2 : g3;
            // candidate v (even): components (x,y); candidate v+1 (odd): (z,w)
            const float sq0 = fmaf(fmaf(-2.0f, d[v],     cnorm), g.x, g.y);
            const float sq1 = fmaf(fmaf(-2.0f, d[v + 1], cnorm), g.z, g.w);
            const unsigned key0 = (__float_as_uint(sq0) & KEY_MASK) |
                                  (unsigned)(base + 8 * h + v);
            const unsigned key1 = (__float_as_uint(sq1) & KEY_MASK) |
                                  (unsigned)(base + 8 * h + v + 1);
            if (__any(umin32(key0, key1) < s[7])) {   // uniform skip once stable
                unsigned cur = key0;
#pragma unroll
                for (int i = 0; i < 8; ++i) {
                    const unsigned lo = umin32(s[i], cur);
                    cur = umax32(s[i], cur);
                    s[i] = lo;
                }
                cur = key1;
#pragma unroll
                for (int i = 0; i < 8; ++i) {
                    const unsigned lo = umin32(s[i], cur);
                    cur = umax32(s[i], cur);
                    s[i] = lo;
                }
            }
        }
    }

    // Merge the two anchor halves per query: smallest-8 of two sorted-8 lists is
    // elementwise min(a_i, b_{7-i}) (first stage of a bitonic 16-merge); the
    // result needn't be sorted — softmax is permutation invariant.
    unsigned p[8];
#pragma unroll
    for (int i = 0; i < 8; ++i)
        p[i] = (unsigned)__shfl_xor((int)s[i], 16, 32);

    unsigned mkey[8];
#pragma unroll
    for (int i = 0; i < 8; ++i) mkey[i] = umin32(s[i], p[7 - i]);

    // Lanes 0-15: exact-distance recompute + softmax for query qn, stash to LDS.
    if (h == 0) {
        float sqv[8];
        int   idv[8];
#pragma unroll
        for (int k = 0; k < 8; ++k) {
            const int idx = (int)(mkey[k] & IDX_MASK);
            idv[k] = idx;
            const float4 pw = pos4[idx];
            const float2 g  = ut[idx];           // g.x = 1/(w^2+eps)
            const float dx = cx - pw.x, dy = cy - pw.y, dz = cz - pw.z;
            sqv[k] = (dx * dx + dy * dy + dz * dz) * g.x;
        }
        float mn = sqv[0];
#pragma unroll
        for (int k = 1; k < 8; ++k) mn = fminf(mn, sqv[k]);
        float e[8], sum = 0.0f;
#pragma unroll
        for (int k = 0; k < 8; ++k) {
            e[k] = __expf((mn - sqv[k]) * INV_TEMP);
            sum += e[k];
        }
        const float inv = 1.0f / sum;
#pragma unroll
        for (int k = 0; k < 8; ++k) {
            lds_w[wave][qn][k] = e[k] * inv;
            lds_i[wave][qn][k] = idv[k];
        }
    }
    __syncthreads();

    // All 32 lanes: gather features (L2-resident) and write output.
    // lane -> (query q2 = lane>>1, feature half h2 = lane&1 covering 32 floats)
    const int q2 = lane >> 1;
    const int h2 = lane & 1;
    const int gq = qbase + q2;

    float4 acc[8];
#pragma unroll
    for (int j = 0; j < 8; ++j) { acc[j].x = 0.f; acc[j].y = 0.f; acc[j].z = 0.f; acc[j].w = 0.f; }

#pragma unroll
    for (int k = 0; k < 8; ++k) {
        const float wk  = lds_w[wave][q2][k];
        const int   idx = lds_i[wave][q2][k];
        const float4* fr = (const float4*)(feat + (size_t)idx * 64) + h2 * 8;
#pragma unroll
        for (int j = 0; j < 8; ++j) {
            const float4 f = fr[j];
            acc[j].x = fmaf(wk, f.x, acc[j].x);
            acc[j].y = fmaf(wk, f.y, acc[j].y);
            acc[j].z = fmaf(wk, f.z, acc[j].z);
            acc[j].w = fmaf(wk, f.w, acc[j].w);
        }
    }

    float4* op = (float4*)(out + (size_t)gq * 64) + h2 * 8;
#pragma unroll
    for (int j = 0; j < 8; ++j) op[j] = acc[j];
}

extern "C" void kernel_launch(void* const* d_in, const int* in_sizes, int n_in,
                              void* d_out, int out_size, void* d_ws, size_t ws_size,
                              hipStream_t stream) {
    const float* coords    = (const float*)d_in[0];
    const float* positions = (const float*)d_in[1];
    const float* weights   = (const float*)d_in[2];
    const float* features  = (const float*)d_in[3];
    float* out = (float*)d_out;

    const int B = in_sizes[0] / 3;
    const int N = in_sizes[2];

    // Scratch layout: pos4 float4[N] (16B each), then ut float2[N] (8B each).
    float4* pos4 = (float4*)d_ws;
    float2* ut   = (float2*)((char*)d_ws + (size_t)N * 16);

    bake_kernel<<<(N + 255) / 256, 256, 0, stream>>>(positions, weights, pos4, ut, N);

    // 256 threads = 8 waves; each wave owns 16 queries -> 128 queries per block.
    knn_interp_kernel<<<B / 128, 256, 0, stream>>>(
        coords, pos4, features, ut, out, N);
}